// ClusteringAffinity_1520418423399
// MI455X (gfx1250) — compile-verified
//
#include <hip/hip_runtime.h>

// Problem constants (from reference setup)
#define B_ROWS   1024
#define D_DIM    1024
#define C_CLUST  1000
#define MC       4000            // C*K
#define OUT_COLS (C_CLUST + 1)
#define SIGMA_INV 0.1f

// fw tiling: 32x32 output tile per wave
#define FW_COL_TILES (MC / 32)                  // 125
#define FW_ROW_TILES (B_ROWS / 32)              // 32
#define FW_WAVES     (FW_ROW_TILES * FW_COL_TILES)  // 4000
#define FW_BLOCKS    (FW_WAVES / 8)             // 500

// ww upper-triangle tiling: 32x32 tiles
#define NTR      (MC / 32)                      // 125 tile rows/cols
#define NT_UPPER (NTR * (NTR + 1) / 2)          // 7875 upper tiles
#define WPB      8                              // waves per block
#define NB3      ((NT_UPPER + WPB - 1) / WPB)   // 985 blocks

// ws layout (floats)
#define WSQ_OFF  0          // 4000 floats: ||w_j||^2
#define FSQ_OFF  4096       // 1024 floats: ||f_b||^2
#define PART_OFF 6144       // 2*NB3 floats: per-block (Sx, Sx2)

typedef __attribute__((ext_vector_type(2))) float v2f;
typedef __attribute__((ext_vector_type(8))) float v8f;

static __device__ __forceinline__ v8f wmma_f32(v2f a, v2f b, v8f c) {
    return __builtin_amdgcn_wmma_f32_16x16x4_f32(false, a, false, b,
                                                 (short)0, c, false, false);
}

// ---------------------------------------------------------------------------
// Kernel 1: row squared-norms for w (rows 0..3999) and f (rows 4000..5023)
// One wave32 per row.
// ---------------------------------------------------------------------------
__global__ __launch_bounds__(256) void rowsq_kernel(const float* __restrict__ f,
                                                    const float* __restrict__ w,
                                                    float* __restrict__ ws) {
    int wave = (blockIdx.x * blockDim.x + threadIdx.x) >> 5;
    int lane = threadIdx.x & 31;
    if (wave >= MC + B_ROWS) return;
    const float* row = (wave < MC) ? (w + (size_t)wave * D_DIM)
                                   : (f + (size_t)(wave - MC) * D_DIM);
    float s = 0.0f;
    #pragma unroll 8
    for (int i = lane; i < D_DIM; i += 32) {
        float v = row[i];
        s = fmaf(v, v, s);
    }
    #pragma unroll
    for (int m = 16; m >= 1; m >>= 1) s += __shfl_xor(s, m, 32);
    if (lane == 0) {
        if (wave < MC) ws[WSQ_OFF + wave] = s;
        else           ws[FSQ_OFF + (wave - MC)] = s;
    }
}

// ---------------------------------------------------------------------------
// Kernel 2: fw = f @ w^T via V_WMMA_F32_16X16X4_F32, fused distance epilogue.
// One wave computes a 32x32 output tile (4 accumulators), register
// double-buffered K loop (prefetch step k+4 while issuing WMMAs for step k).
// ---------------------------------------------------------------------------
__global__ __launch_bounds__(256) void fw_kernel(const float* __restrict__ f,
                                                 const float* __restrict__ w,
                                                 const float* __restrict__ ws,
                                                 float* __restrict__ out) {
    int waveId = blockIdx.x * 8 + (threadIdx.x >> 5);
    int lane   = threadIdx.x & 31;
    int half   = lane >> 4;        // f32 16x4 A layout: lanes16-31 hold K+2/K+3
    int l15    = lane & 15;

    int rowBase = (waveId / FW_COL_TILES) * 32;
    int colBase = (waveId % FW_COL_TILES) * 32;

    const float* aR0 = f + (size_t)(rowBase + l15) * D_DIM + 2 * half;
    const float* aR1 = aR0 + (size_t)16 * D_DIM;
    const float* bR0 = w + (size_t)(colBase + l15) * D_DIM + 2 * half;
    const float* bR1 = bR0 + (size_t)16 * D_DIM;

    v8f acc00 = {}, acc01 = {}, acc10 = {}, acc11 = {};
    v2f a0 = *(const v2f*)aR0;
    v2f a1 = *(const v2f*)aR1;
    v2f b0 = *(const v2f*)bR0;
    v2f b1 = *(const v2f*)bR1;

    #pragma unroll 4
    for (int kk = 4; kk < D_DIM; kk += 4) {
        v2f na0 = *(const v2f*)(aR0 + kk);
        v2f na1 = *(const v2f*)(aR1 + kk);
        v2f nb0 = *(const v2f*)(bR0 + kk);
        v2f nb1 = *(const v2f*)(bR1 + kk);
        acc00 = wmma_f32(a0, b0, acc00);
        acc01 = wmma_f32(a0, b1, acc01);
        acc10 = wmma_f32(a1, b0, acc10);
        acc11 = wmma_f32(a1, b1, acc11);
        a0 = na0; a1 = na1; b0 = nb0; b1 = nb1;
    }
    acc00 = wmma_f32(a0, b0, acc00);
    acc01 = wmma_f32(a0, b1, acc01);
    acc10 = wmma_f32(a1, b0, acc10);
    acc11 = wmma_f32(a1, b1, acc11);

    const float* wsq = ws + WSQ_OFF;
    const float* fsq = ws + FSQ_OFF;
    int col0 = colBase + l15;
    int col1 = col0 + 16;
    float wsq0 = wsq[col0];
    float wsq1 = wsq[col1];

    #pragma unroll
    for (int r = 0; r < 8; ++r) {
        int row0 = rowBase + r + 8 * half;   // C/D layout: lanes16-31 -> M+8
        int row1 = row0 + 16;
        float fs0 = fsq[row0];
        float fs1 = fsq[row1];
        float e00 = __expf(-fmaxf(fs0 + wsq0 - 2.0f * acc00[r], 0.0f) * SIGMA_INV);
        float e01 = __expf(-fmaxf(fs0 + wsq1 - 2.0f * acc01[r], 0.0f) * SIGMA_INV);
        float e10 = __expf(-fmaxf(fs1 + wsq0 - 2.0f * acc10[r], 0.0f) * SIGMA_INV);
        float e11 = __expf(-fmaxf(fs1 + wsq1 - 2.0f * acc11[r], 0.0f) * SIGMA_INV);
        // max over K=4 consecutive columns (lane groups of 4, K-group aligned)
        e00 = fmaxf(e00, __shfl_xor(e00, 1, 32)); e00 = fmaxf(e00, __shfl_xor(e00, 2, 32));
        e01 = fmaxf(e01, __shfl_xor(e01, 1, 32)); e01 = fmaxf(e01, __shfl_xor(e01, 2, 32));
        e10 = fmaxf(e10, __shfl_xor(e10, 1, 32)); e10 = fmaxf(e10, __shfl_xor(e10, 2, 32));
        e11 = fmaxf(e11, __shfl_xor(e11, 1, 32)); e11 = fmaxf(e11, __shfl_xor(e11, 2, 32));
        if ((lane & 3) == 0) {
            out[(size_t)row0 * OUT_COLS + (col0 >> 2)] = e00;
            out[(size_t)row0 * OUT_COLS + (col1 >> 2)] = e01;
            out[(size_t)row1 * OUT_COLS + (col0 >> 2)] = e10;
            out[(size_t)row1 * OUT_COLS + (col1 >> 2)] = e11;
        }
    }
}

// ---------------------------------------------------------------------------
// Kernel 3: upper-triangular 32x32 tiles of ww = w @ w^T; per-block
// deterministic partial sums of Sx = sum(x), Sx2 = sum(x^2) over i<=j, where
// x = max(|wi|^2 + |wj|^2 - 2<wi,wj>, 0).
// ---------------------------------------------------------------------------
__global__ __launch_bounds__(256) void ww_kernel(const float* __restrict__ w,
                                                 const float* __restrict__ ws,
                                                 float* __restrict__ part) {
    __shared__ float sSx[WPB];
    __shared__ float sSx2[WPB];
    int waveInBlk = threadIdx.x >> 5;
    int lane      = threadIdx.x & 31;
    int t         = blockIdx.x * WPB + waveInBlk;

    float sx = 0.0f, sx2 = 0.0f;
    if (t < NT_UPPER) {                         // wave-uniform branch
        // decode linear upper-tri tile index -> (ti, tj), ti <= tj
        const int n = NTR;
        float disc = (float)((2 * n + 1) * (2 * n + 1)) - 8.0f * (float)t;
        int ti = (int)(((float)(2 * n + 1) - sqrtf(disc)) * 0.5f);
        if (ti < 0) ti = 0;
        if (ti > n - 1) ti = n - 1;
        while (ti > 0 && (ti * (2 * n - ti + 1)) / 2 > t) --ti;
        while (((ti + 1) * (2 * n - ti)) / 2 <= t) ++ti;
        int tj = ti + (t - (ti * (2 * n - ti + 1)) / 2);

        int half = lane >> 4;
        int l15  = lane & 15;
        int rowBase = ti * 32;
        int colBase = tj * 32;

        const float* aR0 = w + (size_t)(rowBase + l15) * D_DIM + 2 * half;
        const float* aR1 = aR0 + (size_t)16 * D_DIM;
        const float* bR0 = w + (size_t)(colBase + l15) * D_DIM + 2 * half;
        const float* bR1 = bR0 + (size_t)16 * D_DIM;

        v8f acc00 = {}, acc01 = {}, acc10 = {}, acc11 = {};
        v2f a0 = *(const v2f*)aR0;
        v2f a1 = *(const v2f*)aR1;
        v2f b0 = *(const v2f*)bR0;
        v2f b1 = *(const v2f*)bR1;

        #pragma unroll 4
        for (int kk = 4; kk < D_DIM; kk += 4) {
            v2f na0 = *(const v2f*)(aR0 + kk);
            v2f na1 = *(const v2f*)(aR1 + kk);
            v2f nb0 = *(const v2f*)(bR0 + kk);
            v2f nb1 = *(const v2f*)(bR1 + kk);
            acc00 = wmma_f32(a0, b0, acc00);
            acc01 = wmma_f32(a0, b1, acc01);
            acc10 = wmma_f32(a1, b0, acc10);
            acc11 = wmma_f32(a1, b1, acc11);
            a0 = na0; a1 = na1; b0 = nb0; b1 = nb1;
        }
        acc00 = wmma_f32(a0, b0, acc00);
        acc01 = wmma_f32(a0, b1, acc01);
        acc10 = wmma_f32(a1, b0, acc10);
        acc11 = wmma_f32(a1, b1, acc11);

        const float* wsq = ws + WSQ_OFF;
        int j0 = colBase + l15;
        int j1 = j0 + 16;
        float wsqj0 = wsq[j0];
        float wsqj1 = wsq[j1];
        #pragma unroll
        for (int r = 0; r < 8; ++r) {
            int i0 = rowBase + r + 8 * half;
            int i1 = i0 + 16;
            float wsqi0 = wsq[i0];
            float wsqi1 = wsq[i1];
            float x00 = fmaxf(wsqi0 + wsqj0 - 2.0f * acc00[r], 0.0f);
            float x01 = fmaxf(wsqi0 + wsqj1 - 2.0f * acc01[r], 0.0f);
            float x10 = fmaxf(wsqi1 + wsqj0 - 2.0f * acc10[r], 0.0f);
            float x11 = fmaxf(wsqi1 + wsqj1 - 2.0f * acc11[r], 0.0f);
            if (i0 <= j0) { sx += x00; sx2 = fmaf(x00, x00, sx2); }
            if (i0 <= j1) { sx += x01; sx2 = fmaf(x01, x01, sx2); }
            if (i1 <= j0) { sx += x10; sx2 = fmaf(x10, x10, sx2); }
            if (i1 <= j1) { sx += x11; sx2 = fmaf(x11, x11, sx2); }
        }
    }
    #pragma unroll
    for (int m = 16; m >= 1; m >>= 1) {
        sx  += __shfl_xor(sx,  m, 32);
        sx2 += __shfl_xor(sx2, m, 32);
    }
    if (lane == 0) { sSx[waveInBlk] = sx; sSx2[waveInBlk] = sx2; }
    __syncthreads();
    if (threadIdx.x == 0) {
        float a = 0.0f, b = 0.0f;
        #pragma unroll
        for (int i = 0; i < WPB; ++i) { a += sSx[i]; b += sSx2[i]; }
        part[(size_t)blockIdx.x * 2 + 0] = a;
        part[(size_t)blockIdx.x * 2 + 1] = b;
    }
}

// ---------------------------------------------------------------------------
// Kernel 4: reduce partials (fixed order -> deterministic), compute rw,
// broadcast into out[:, C].
// rw = denom * sum_{i<=j}(x - mu)^2 = denom*(Sx2 - 2*mu*Sx + Nu*mu^2)
// ---------------------------------------------------------------------------
__global__ __launch_bounds__(256) void finalize_kernel(const float* __restrict__ part,
                                                       float* __restrict__ out) {
    __shared__ float sA[256];
    __shared__ float sB[256];
    __shared__ float sRw;
    float a = 0.0f, b = 0.0f;
    for (int i = threadIdx.x; i < NB3; i += 256) {
        a += part[(size_t)i * 2 + 0];
        b += part[(size_t)i * 2 + 1];
    }
    sA[threadIdx.x] = a;
    sB[threadIdx.x] = b;
    __syncthreads();
    for (int s = 128; s > 0; s >>= 1) {
        if (threadIdx.x < (unsigned)s) {
            sA[threadIdx.x] += sA[threadIdx.x + s];
            sB[threadIdx.x] += sB[threadIdx.x + s];
        }
        __syncthreads();
    }
    if (threadIdx.x == 0) {
        double Sx  = (double)sA[0];
        double Sx2 = (double)sB[0];
        double denom = 2.0 / ((double)MC * (double)MC - (double)MC);
        double mu = denom * Sx;
        double Nu = (double)MC * (double)(MC + 1) * 0.5;   // upper incl diag
        double rw = denom * (Sx2 - 2.0 * mu * Sx + Nu * mu * mu);
        sRw = (float)rw;
    }
    __syncthreads();
    float rw = sRw;
    for (int bi = threadIdx.x; bi < B_ROWS; bi += 256)
        out[(size_t)bi * OUT_COLS + C_CLUST] = rw;
}

// ---------------------------------------------------------------------------
extern "C" void kernel_launch(void* const* d_in, const int* in_sizes, int n_in,
                              void* d_out, int out_size, void* d_ws, size_t ws_size,
                              hipStream_t stream) {
    (void)in_sizes; (void)n_in; (void)out_size; (void)ws_size;
    const float* f       = (const float*)d_in[0];   // [1024, 1024]
    const float* centers = (const float*)d_in[1];   // [1000, 4, 1024] -> [4000, 1024]
    float* out = (float*)d_out;                     // [1024, 1001]
    float* ws  = (float*)d_ws;
    float* part = ws + PART_OFF;

    rowsq_kernel<<<(MC + B_ROWS) / 8, 256, 0, stream>>>(f, centers, ws);
    fw_kernel<<<FW_BLOCKS, 256, 0, stream>>>(f, centers, ws, out);
    ww_kernel<<<NB3, 256, 0, stream>>>(centers, ws, part);
    finalize_kernel<<<1, 256, 0, stream>>>(part, out);
}